// TopKGate_56599079027007
// MI455X (gfx1250) — compile-verified
//
#include <hip/hip_runtime.h>

// MoE top-k gating for MI455X (gfx1250, wave32).
//   logits[N,64] = X[N,1024] @ W[64,1024]^T + b   via V_WMMA_F32_16X16X4_F32
//   probs = softmax(logits); top-8 indices + softmax gates.
// Memory-roofline: X (134 MB) read once, non-temporally; W (256 KB) stays in L2.
// Each wave: 32 rows x 64 experts = 2x4 tiles of 16x16 f32 accumulators.

typedef float v2f __attribute__((ext_vector_type(2)));
typedef float v4f __attribute__((ext_vector_type(4)));
typedef float v8f __attribute__((ext_vector_type(8)));

namespace {
constexpr int Dk    = 1024;          // reduction dim
constexpr int Ex    = 64;            // experts
constexpr int TK    = 8;             // top-k
constexpr int WAVES = 4;             // waves per block
constexpr int RPW   = 32;            // rows per wave (2 M-tiles of 16)
constexpr int RPB   = WAVES * RPW;   // 128 rows per block
constexpr int LSTR  = 67;            // LDS row stride (67 % 64 == 3, coprime -> conflict-free)
}

__global__ __launch_bounds__(WAVES * 32) void topk_gate_kernel(
    const float* __restrict__ inp,   // [N, Dk]
    const float* __restrict__ Wm,    // [Ex, Dk]
    const float* __restrict__ bias,  // [Ex]
    float* __restrict__ out_idx,     // [N, TK]  (indices stored as float values)
    float* __restrict__ out_gate,    // [N, TK]
    float* __restrict__ out_prob)    // [N, Ex]
{
  __shared__ float lds[WAVES][RPW][LSTR];

  const int lane = threadIdx.x & 31;
  const int wave = threadIdx.x >> 5;
  const int rw   = lane & 15;          // row-in-tile (A) / expert-in-tile (B)
  const int kh   = (lane >> 4) * 2;    // K sub-offset for 16x4 f32 fragments
  const int rowBase = blockIdx.x * RPB + wave * RPW;

  // Per-lane base pointers; every fragment load is one b64.
  // A 16x4 f32 layout: lanes 0-15 -> {K=0,K=1}, lanes 16-31 -> {K=2,K=3}, row = lane&15.
  const float* a0p = inp + (size_t)(rowBase + rw) * Dk + kh;
  const float* a1p = a0p + (size_t)16 * Dk;
  const float* b0p = Wm + (size_t)rw * Dk + kh;   // expert tile t adds t*16*Dk

  v8f acc[2][4];
  #pragma unroll
  for (int m = 0; m < 2; ++m)
    #pragma unroll
    for (int t = 0; t < 4; ++t)
      acc[m][t] = v8f{};

  #pragma unroll 2
  for (int k0 = 0; k0 < Dk; k0 += 4) {
    // X: streamed, zero reuse -> non-temporal (TH=NT), keep L2 for W.
    v2f a0 = __builtin_nontemporal_load((const v2f*)(a0p + k0));
    v2f a1 = __builtin_nontemporal_load((const v2f*)(a1p + k0));
    v2f bf[4];
    #pragma unroll
    for (int t = 0; t < 4; ++t)
      bf[t] = *(const v2f*)(b0p + (size_t)t * 16 * Dk + k0);
    #pragma unroll
    for (int t = 0; t < 4; ++t) {
      acc[0][t] = __builtin_amdgcn_wmma_f32_16x16x4_f32(
          false, a0, false, bf[t], (short)0, acc[0][t], false, false);
      acc[1][t] = __builtin_amdgcn_wmma_f32_16x16x4_f32(
          false, a1, false, bf[t], (short)0, acc[1][t], false, false);
    }
  }

  // Add bias, transpose 16x16 f32 C layout (VGPR v -> row, lane -> expert) via LDS.
  float bv[4];
  #pragma unroll
  for (int t = 0; t < 4; ++t) bv[t] = bias[t * 16 + rw];

  #pragma unroll
  for (int m = 0; m < 2; ++m)
    #pragma unroll
    for (int t = 0; t < 4; ++t)
      #pragma unroll
      for (int v = 0; v < 8; ++v)
        lds[wave][m * 16 + (lane >> 4) * 8 + v][t * 16 + rw] = acc[m][t][v] + bv[t];

  __syncthreads();

  // ---- Epilogue: one lane == one full row of 64 logits ----
  const int gr = rowBase + lane;
  float* row = &lds[wave][lane][0];

  float mx = -3.402823466e38f;
  for (int j = 0; j < Ex; ++j) mx = fmaxf(mx, row[j]);

  float s = 0.0f;
  for (int j = 0; j < Ex; ++j) {
    float e = __expf(row[j] - mx);
    row[j] = e;                       // overwrite logits with exp(logit - max)
    s += e;
  }
  const float inv = 1.0f / s;

  // Full softmax probabilities: [N, 64], 16 x v4 non-temporal stores per row.
  v4f* pp = (v4f*)(out_prob + (size_t)gr * Ex);
  for (int j = 0; j < Ex / 4; ++j) {
    v4f pv = { row[4*j+0] * inv, row[4*j+1] * inv, row[4*j+2] * inv, row[4*j+3] * inv };
    __builtin_nontemporal_store(pv, pp + j);
  }

  // Top-8 by (value desc, index asc) — eligibility mask instead of mutating.
  // exp() is monotonic, so ranking exp-values == ranking logits, and since
  // top-1 IS the row max, gates = tv[k] / sum(tv) is the exact stabilized softmax.
  float tv[TK];
  int   ti[TK];
  float pvv = 3.402823466e38f;
  int   pii = -1;
  for (int k = 0; k < TK; ++k) {
    float bm = -1.0f;
    int   bi = 0;
    for (int j = 0; j < Ex; ++j) {
      float vj = row[j];
      bool elig = (vj < pvv) || (vj == pvv && j > pii);
      if (elig && vj > bm) { bm = vj; bi = j; }
    }
    tv[k] = bm; ti[k] = bi;
    pvv = bm; pii = bi;
  }

  float gs = 0.0f;
  #pragma unroll
  for (int k = 0; k < TK; ++k) gs += tv[k];
  const float gi = 1.0f / gs;

  v4f i0 = { (float)ti[0], (float)ti[1], (float)ti[2], (float)ti[3] };
  v4f i1 = { (float)ti[4], (float)ti[5], (float)ti[6], (float)ti[7] };
  v4f g0 = { tv[0]*gi, tv[1]*gi, tv[2]*gi, tv[3]*gi };
  v4f g1 = { tv[4]*gi, tv[5]*gi, tv[6]*gi, tv[7]*gi };
  __builtin_nontemporal_store(i0, (v4f*)(out_idx  + (size_t)gr * TK) + 0);
  __builtin_nontemporal_store(i1, (v4f*)(out_idx  + (size_t)gr * TK) + 1);
  __builtin_nontemporal_store(g0, (v4f*)(out_gate + (size_t)gr * TK) + 0);
  __builtin_nontemporal_store(g1, (v4f*)(out_gate + (size_t)gr * TK) + 1);
}

extern "C" void kernel_launch(void* const* d_in, const int* in_sizes, int n_in,
                              void* d_out, int out_size, void* d_ws, size_t ws_size,
                              hipStream_t stream) {
  (void)n_in; (void)out_size; (void)d_ws; (void)ws_size;
  const float* inp  = (const float*)d_in[0];   // [N, 1024]
  const float* Wm   = (const float*)d_in[1];   // [64, 1024]
  const float* bias = (const float*)d_in[2];   // [64]
  const int N = in_sizes[0] / Dk;              // 32768

  float* out      = (float*)d_out;
  float* out_idx  = out;                                // [N, 8] indices (as float)
  float* out_gate = out + (size_t)N * TK;               // [N, 8]
  float* out_prob = out + (size_t)2 * N * TK;           // [N, 64]

  dim3 grid(N / RPB);          // 256 blocks
  dim3 block(WAVES * 32);      // 128 threads = 4 waves
  hipLaunchKernelGGL(topk_gate_kernel, grid, block, 0, stream,
                     inp, Wm, bias, out_idx, out_gate, out_prob);
}